// DomainSepLoss_88235808129500
// MI455X (gfx1250) — compile-verified
//
#include <hip/hip_runtime.h>

typedef float v2f __attribute__((ext_vector_type(2)));
typedef float v8f __attribute__((ext_vector_type(8)));

#define B_ 16
#define N_ 8192
#define C_ 128
#define NC_ (N_ * C_)

// ---- workspace layout (float units) ----
#define WS_GRAM    0                              // 2*16*128*128 = 524288
#define WS_NORMSQ  (WS_GRAM + 2 * B_ * C_ * C_)   // 4*16*128 = 8192
#define WS_ACC     (WS_NORMSQ + 4 * B_ * C_)      // 16 scalars
#define WS_POS     (WS_ACC + 16)                  // 2*8192 ints (zeroed as floats)
#define WS_ZERO_N  (WS_POS + 2 * N_)
#define WS_MEAN    WS_ZERO_N                      // 4*NC_
#define WS_ORDER   (WS_MEAN + 4 * NC_)            // 2*8192 ints

// acc indices: 0 sim, 1 diff0, 2 diff1, 3 recon0, 4 recon1, 5 gnn, 6 cnt0, 7 cnt1

// -------------------- zero init --------------------
__global__ void zero_ws_k(float* ws) {
    int i = blockIdx.x * 256 + threadIdx.x;
    if (i < WS_ZERO_N) ws[i] = 0.f;
}

// -------------------- batch means over axis 0 --------------------
__global__ void mean_k(const float* __restrict__ a, const float* __restrict__ b,
                       const float* __restrict__ c, const float* __restrict__ d,
                       float* __restrict__ mean) {
    int idx = blockIdx.x * 256 + threadIdx.x;           // 0..NC_-1
    int t = blockIdx.y;
    const float* s = (t == 0) ? a : (t == 1) ? b : (t == 2) ? c : d;
    float sum = 0.f;
#pragma unroll
    for (int bb = 0; bb < B_; ++bb) sum += s[(size_t)bb * NC_ + idx];
    mean[(size_t)t * NC_ + idx] = sum * (1.f / 16.f);
}

// -------------------- centered column sum-of-squares --------------------
__global__ void normsq_k(const float* __restrict__ a, const float* __restrict__ b,
                         const float* __restrict__ c, const float* __restrict__ d,
                         const float* __restrict__ mean, float* __restrict__ normsq) {
    int t = blockIdx.z, bIdx = blockIdx.y, slice = blockIdx.x;
    const float* s = (t == 0) ? a : (t == 1) ? b : (t == 2) ? c : d;
    const float* m = mean + (size_t)t * NC_;
    int cc = threadIdx.x & 127, h = threadIdx.x >> 7;
    int n0 = slice * 1024 + h * 512;
    float acc = 0.f;
    for (int n = n0; n < n0 + 512; ++n) {
        float v = s[((size_t)bIdx * N_ + n) * C_ + cc] - m[(size_t)n * C_ + cc];
        acc += v * v;
    }
    __shared__ float sm[256];
    sm[threadIdx.x] = acc;
    __syncthreads();
    if (threadIdx.x < 128)
        atomicAdd(&normsq[(t * B_ + bIdx) * C_ + cc], sm[cc] + sm[cc + 128]);
}

// -------------------- WMMA Gram: G[c,d] = sum_n Ac[n,c]*Bc[n,d] --------------------
// LDS tiles stored in paired-K layout: element (k,c) at dword (k>>1)*256 + 2c + (k&1),
// so every 16x4 A / 4x16 B f32 fragment is one aligned ds_load_b64 into an even VGPR pair.
__global__ void gram_k(const float* __restrict__ hs, const float* __restrict__ hp,
                       const float* __restrict__ ls, const float* __restrict__ lp,
                       const float* __restrict__ mean, float* __restrict__ gram) {
    int pair = blockIdx.z, bIdx = blockIdx.y, slice = blockIdx.x;
    const float* A  = pair ? ls : hs;
    const float* Bm = pair ? lp : hp;
    const float* mA = mean + (size_t)(pair ? 2 : 0) * NC_;
    const float* mB = mean + (size_t)(pair ? 3 : 1) * NC_;

    __shared__ float As[32 * 128];   // paired-K layout
    __shared__ float Bs[32 * 128];

    int tid = threadIdx.x;
    int wave = tid >> 5, lane = tid & 31;
    int h = lane >> 4, l4 = lane & 15;

    v8f acc[8] = {};
    size_t gbase = (size_t)bIdx * N_ * C_;

    for (int chunk = 0; chunk < 32; ++chunk) {
        int nbase = slice * 1024 + chunk * 32;
        // cooperative centered tile load: 512 row-pair/c-quad units per matrix,
        // each unit = 2 consecutive rows x 4 cols -> two b128 LDS stores (2x2 transposed blocks)
#pragma unroll
        for (int i = 0; i < 2; ++i) {
            int u = tid + i * 256;           // 0..511
            int pr = u >> 5;                 // row pair 0..15
            int c  = (u & 31) * 4;           // col quad
            size_t g0 = gbase + (size_t)(nbase + pr * 2) * C_ + c;
            size_t m0 = (size_t)(nbase + pr * 2) * C_ + c;
            int base = pr * 256 + c * 2;     // LDS dword base

            float4 xa0 = *(const float4*)&A[g0];
            float4 xa1 = *(const float4*)&A[g0 + C_];
            float4 ma0 = *(const float4*)&mA[m0];
            float4 ma1 = *(const float4*)&mA[m0 + C_];
            float4 r0, r1;
            r0.x = xa0.x - ma0.x; r0.y = xa0.y - ma0.y; r0.z = xa0.z - ma0.z; r0.w = xa0.w - ma0.w;
            r1.x = xa1.x - ma1.x; r1.y = xa1.y - ma1.y; r1.z = xa1.z - ma1.z; r1.w = xa1.w - ma1.w;
            *(float4*)&As[base + 0] = make_float4(r0.x, r1.x, r0.y, r1.y);
            *(float4*)&As[base + 4] = make_float4(r0.z, r1.z, r0.w, r1.w);

            float4 xb0 = *(const float4*)&Bm[g0];
            float4 xb1 = *(const float4*)&Bm[g0 + C_];
            float4 mb0 = *(const float4*)&mB[m0];
            float4 mb1 = *(const float4*)&mB[m0 + C_];
            r0.x = xb0.x - mb0.x; r0.y = xb0.y - mb0.y; r0.z = xb0.z - mb0.z; r0.w = xb0.w - mb0.w;
            r1.x = xb1.x - mb1.x; r1.y = xb1.y - mb1.y; r1.z = xb1.z - mb1.z; r1.w = xb1.w - mb1.w;
            *(float4*)&Bs[base + 0] = make_float4(r0.x, r1.x, r0.y, r1.y);
            *(float4*)&Bs[base + 4] = make_float4(r0.z, r1.z, r0.w, r1.w);
        }
        __syncthreads();

#pragma unroll
        for (int koff = 0; koff < 32; koff += 4) {
            // A fragment: lanes 0-15 K={koff,koff+1}, lanes 16-31 K={koff+2,koff+3}
            int krow = (koff >> 1) + h;      // paired-K row index (k0 = koff + 2h, even)
            v2f afrag = *(const v2f*)&As[krow * 256 + (wave * 16 + l4) * 2];
#pragma unroll
            for (int di = 0; di < 8; ++di) {
                v2f bfrag = *(const v2f*)&Bs[krow * 256 + (di * 16 + l4) * 2];
                acc[di] = __builtin_amdgcn_wmma_f32_16x16x4_f32(
                    false, afrag, false, bfrag, (short)0, acc[di], false, false);
            }
        }
        __syncthreads();
    }

    // write out: D VGPR r holds M = r + 8*h, N = l4 within each 16x16 tile
    float* g = gram + (size_t)(pair * B_ + bIdx) * C_ * C_;
#pragma unroll
    for (int di = 0; di < 8; ++di) {
#pragma unroll
        for (int r = 0; r < 8; ++r) {
            int cRow = wave * 16 + r + 8 * h;
            int dCol = di * 16 + l4;
            atomicAdd(&g[cRow * C_ + dCol], acc[di][r]);
        }
    }
}

// -------------------- normalize Gram, sum corr^2 --------------------
__global__ void gramfin_k(const float* __restrict__ gram, const float* __restrict__ normsq,
                          float* __restrict__ acc_) {
    int pb = blockIdx.x;
    int pair = pb >> 4, bIdx = pb & 15;
    const float* g = gram + (size_t)pb * C_ * C_;
    const float* nsA = normsq + ((pair ? 2 : 0) * B_ + bIdx) * C_;
    const float* nsB = normsq + ((pair ? 3 : 1) * B_ + bIdx) * C_;
    float s = 0.f;
    for (int i = threadIdx.x; i < C_ * C_; i += 256) {
        int cRow = i >> 7, dCol = i & 127;
        float na = fmaxf(sqrtf(nsA[cRow]), 1e-12f);
        float nb = fmaxf(sqrtf(nsB[dCol]), 1e-12f);
        float corr = g[i] / (na * nb);
        s += corr * corr;
    }
    __shared__ float sm[256];
    sm[threadIdx.x] = s;
    __syncthreads();
    for (int off = 128; off > 0; off >>= 1) {
        if (threadIdx.x < off) sm[threadIdx.x] += sm[threadIdx.x + off];
        __syncthreads();
    }
    if (threadIdx.x == 0) atomicAdd(&acc_[1 + pair], sm[0]);
}

// -------------------- Pearson on first 16 flattened rows --------------------
__global__ void pearson_k(const float* __restrict__ x, const float* __restrict__ y,
                          float* __restrict__ acc_) {
    __shared__ float sm[5 * 128];
    int c = threadIdx.x;  // 128 threads
    float loss = 0.f;
    for (int row = 0; row < 16; ++row) {
        float xv = x[row * C_ + c], yv = y[row * C_ + c];
        sm[c] = xv; sm[128 + c] = yv; sm[256 + c] = xv * xv;
        sm[384 + c] = yv * yv; sm[512 + c] = xv * yv;
        __syncthreads();
        for (int off = 64; off > 0; off >>= 1) {
            if (c < off) {
#pragma unroll
                for (int k = 0; k < 5; ++k) sm[k * 128 + c] += sm[k * 128 + c + off];
            }
            __syncthreads();
        }
        if (c == 0) {
            float sx = sm[0], sy = sm[128], sxx = sm[256], syy = sm[384], sxy = sm[512];
            const float n = 128.f;
            float num = sxy - sx * sy / n;
            float den = sqrtf((sxx - sx * sx / n) * (syy - sy * sy / n));
            float r = (den == 0.f) ? 0.f : num / den;
            loss += 1.f - r;
        }
        __syncthreads();
    }
    if (c == 0) acc_[0] = loss / 16.f;
}

// -------------------- recon MSE --------------------
__global__ void recon_k(const float* __restrict__ r0, const float* __restrict__ o0,
                        const float* __restrict__ r1, const float* __restrict__ o1,
                        float* __restrict__ acc_) {
    int pair = blockIdx.y;
    const float4* a = (const float4*)(pair ? r1 : r0);
    const float4* b = (const float4*)(pair ? o1 : o0);
    const int total4 = B_ * NC_ / 4;
    float s = 0.f;
    for (int i = blockIdx.x * blockDim.x + threadIdx.x; i < total4; i += gridDim.x * blockDim.x) {
        float4 av = a[i], bv = b[i];
        float dx = av.x - bv.x, dy = av.y - bv.y, dz = av.z - bv.z, dw = av.w - bv.w;
        s += dx * dx + dy * dy + dz * dz + dw * dw;
    }
    __shared__ float sm[256];
    sm[threadIdx.x] = s;
    __syncthreads();
    for (int off = 128; off > 0; off >>= 1) {
        if (threadIdx.x < off) sm[threadIdx.x] += sm[threadIdx.x + off];
        __syncthreads();
    }
    if (threadIdx.x == 0) atomicAdd(&acc_[3 + pair], sm[0]);
}

// -------------------- stable descending-rank position count --------------------
__global__ void poscount_k(const float* __restrict__ gt, int* __restrict__ pos,
                           float* __restrict__ acc_) {
    int row = blockIdx.y, kslice = blockIdx.x;  // 8 k-slices
    __shared__ float v[N_];
    for (int i = threadIdx.x; i < N_; i += 256) v[i] = gt[row * N_ + i];
    __syncthreads();

    float vj[32];
    int cnt[32];
#pragma unroll
    for (int jj = 0; jj < 32; ++jj) { vj[jj] = v[jj * 256 + threadIdx.x]; cnt[jj] = 0; }

    int k0 = kslice * 1024;
    for (int k = k0; k < k0 + 1024; ++k) {
        float vk = v[k];
#pragma unroll
        for (int jj = 0; jj < 32; ++jj) {
            int j = jj * 256 + threadIdx.x;
            cnt[jj] += ((vk > vj[jj]) || (vk == vj[jj] && k < j)) ? 1 : 0;
        }
    }
#pragma unroll
    for (int jj = 0; jj < 32; ++jj)
        atomicAdd(&pos[row * N_ + jj * 256 + threadIdx.x], cnt[jj]);

    if (kslice == 0) {  // count positives once per row
        int p = 0;
#pragma unroll
        for (int jj = 0; jj < 32; ++jj) p += (vj[jj] > 0.f) ? 1 : 0;
        __shared__ float red[256];
        red[threadIdx.x] = (float)p;
        __syncthreads();
        for (int off = 128; off > 0; off >>= 1) {
            if (threadIdx.x < off) red[threadIdx.x] += red[threadIdx.x + off];
            __syncthreads();
        }
        if (threadIdx.x == 0) acc_[6 + row] = red[0];
    }
}

__global__ void scatter_k(const int* __restrict__ pos, int* __restrict__ order) {
    int row = blockIdx.y;
    int j = blockIdx.x * 256 + threadIdx.x;
    order[row * N_ + pos[row * N_ + j]] = j;
}

// -------------------- masked cosine-embedding sum --------------------
__global__ void cossim_k(const float* __restrict__ sep, const int* __restrict__ order,
                         float* __restrict__ acc_) {
    int i = blockIdx.x;
    int s0 = (int)acc_[6]; if (s0 == 0) s0 = N_;
    int s1 = (int)acc_[7]; if (s1 == 0) s1 = N_;
    int index = s0 < s1 ? s0 : s1;
    if (i >= index) return;
    int o0 = order[i];
    int o1 = order[N_ + i];
    int c = threadIdx.x;  // 128 threads
    float a = sep[(size_t)o0 * C_ + c];
    float b = sep[(size_t)(N_ + o1) * C_ + c];
    __shared__ float sd[128], sa[128], sb[128];
    sd[c] = a * b; sa[c] = a * a; sb[c] = b * b;
    __syncthreads();
    for (int off = 64; off > 0; off >>= 1) {
        if (c < off) { sd[c] += sd[c + off]; sa[c] += sa[c + off]; sb[c] += sb[c + off]; }
        __syncthreads();
    }
    if (c == 0) {
        float cosv = sd[0] / (fmaxf(sqrtf(sa[0]), 1e-8f) * fmaxf(sqrtf(sb[0]), 1e-8f));
        atomicAdd(&acc_[5], 1.f - cosv);
    }
}

// -------------------- final combine --------------------
__global__ void combine_k(const float* __restrict__ acc_, float* __restrict__ out) {
    float sim = acc_[0];
    float denomD = (float)(B_ * C_ * C_);
    float diff = fmaxf(acc_[1] / denomD, 0.f) + fmaxf(acc_[2] / denomD, 0.f);
    float tot = (float)B_ * (float)NC_;
    float recon = acc_[3] / tot + acc_[4] / tot;
    int s0 = (int)acc_[6]; if (s0 == 0) s0 = N_;
    int s1 = (int)acc_[7]; if (s1 == 0) s1 = N_;
    int index = s0 < s1 ? s0 : s1;
    float gnn = acc_[5] / (float)index;
    out[0] = 2.0f * sim + 1.0f * diff + 0.05f * recon + gnn;
}

extern "C" void kernel_launch(void* const* d_in, const int* in_sizes, int n_in,
                              void* d_out, int out_size, void* d_ws, size_t ws_size,
                              hipStream_t stream) {
    const float* hf_feature  = (const float*)d_in[0];
    const float* lf_feature  = (const float*)d_in[1];
    const float* hf_shared   = (const float*)d_in[2];
    const float* lf_shared   = (const float*)d_in[3];
    const float* hf_private  = (const float*)d_in[4];
    const float* lf_private  = (const float*)d_in[5];
    const float* restored_hf = (const float*)d_in[6];
    const float* restored_lf = (const float*)d_in[7];
    const float* sep_node    = (const float*)d_in[8];
    const float* gt_rank     = (const float*)d_in[10];

    float* ws      = (float*)d_ws;
    float* gram    = ws + WS_GRAM;
    float* normsq  = ws + WS_NORMSQ;
    float* acc     = ws + WS_ACC;
    int*   pos     = (int*)(ws + WS_POS);
    float* mean    = ws + WS_MEAN;
    int*   order   = (int*)(ws + WS_ORDER);
    float* out     = (float*)d_out;

    zero_ws_k<<<(WS_ZERO_N + 255) / 256, 256, 0, stream>>>(ws);

    mean_k<<<dim3(NC_ / 256, 4), 256, 0, stream>>>(hf_shared, hf_private, lf_shared, lf_private, mean);

    normsq_k<<<dim3(8, B_, 4), 256, 0, stream>>>(hf_shared, hf_private, lf_shared, lf_private,
                                                 mean, normsq);

    gram_k<<<dim3(8, B_, 2), 256, 0, stream>>>(hf_shared, hf_private, lf_shared, lf_private,
                                               mean, gram);

    gramfin_k<<<32, 256, 0, stream>>>(gram, normsq, acc);

    pearson_k<<<1, 128, 0, stream>>>(hf_shared, lf_shared, acc);

    recon_k<<<dim3(512, 2), 256, 0, stream>>>(restored_hf, hf_feature, restored_lf, lf_feature, acc);

    poscount_k<<<dim3(8, 2), 256, 0, stream>>>(gt_rank, pos, acc);
    scatter_k<<<dim3(N_ / 256, 2), 256, 0, stream>>>(pos, order);
    cossim_k<<<N_, 128, 0, stream>>>(sep_node, order, acc);

    combine_k<<<1, 1, 0, stream>>>(acc, out);
}